// SheafDiffusion_60756607369830
// MI455X (gfx1250) — compile-verified
//
#include <hip/hip_runtime.h>
#include <hip/hip_bf16.h>

typedef __attribute__((ext_vector_type(2))) float v2f;
typedef __attribute__((ext_vector_type(8))) float v8f;

#define G       50000
#define D       3
#define HC      64
#define IN_DIM  256
#define OUT_DIM 40
#define NL      4
#define NNZ     2400000
#define NN      (G * D)      /* 150000 extended nodes */
#define HD      (HC * D)     /* 192 */
#define OUT_PAD 48           /* 40 padded to 3 WMMA tiles */

__device__ __forceinline__ float eluf(float x) {
    return x > 0.f ? x : (__expf(x) - 1.f);
}

// ---------------------------------------------------------------------------
// lin1: h = elu(X[G,256] @ W1[192,256]^T + b1), also h0 = h.
// One wave computes one 16x16 output tile via V_WMMA_F32_16X16X4_F32.
// A fragment layout (16x4 f32): lane&15 = M, lane>>4 selects K-half, v2f = K pair.
// C/D layout: vgpr v, lanes 0-15 -> M=v, lanes 16-31 -> M=v+8; N = lane&15.
// ---------------------------------------------------------------------------
__global__ void lin1_kernel(const float* __restrict__ X,
                            const float* __restrict__ W,
                            const float* __restrict__ bias,
                            float* __restrict__ h,
                            float* __restrict__ h0) {
    int wid  = (int)((blockIdx.x * blockDim.x + threadIdx.x) >> 5);
    int lane = threadIdx.x & 31;
    const int mtiles = G / 16;     // 3125
    const int ntiles = HD / 16;    // 12
    if (wid >= mtiles * ntiles) return;
    int mt = wid / ntiles, nt = wid % ntiles;
    int r0 = mt * 16, c0 = nt * 16;
    int half = lane >> 4;
    int lidx = lane & 15;

    const float* arow = X + (size_t)(r0 + lidx) * IN_DIM;
    const float* brow = W + (size_t)(c0 + lidx) * IN_DIM;

    v8f acc = {};
    for (int k = 0; k < IN_DIM; k += 4) {
        v2f a = *(const v2f*)(arow + k + 2 * half);
        v2f b = *(const v2f*)(brow + k + 2 * half);
        acc = __builtin_amdgcn_wmma_f32_16x16x4_f32(
            false, a, false, b, (short)0, acc, false, false);
    }

    float bv = bias[c0 + lidx];
    #pragma unroll
    for (int v = 0; v < 8; ++v) {
        int row = r0 + v + 8 * half;
        size_t idx = (size_t)row * HD + c0 + lidx;
        float val = eluf(acc[v] + bv);
        h[idx]  = val;
        h0[idx] = val;
    }
}

// ---------------------------------------------------------------------------
// Left stalk-dim mix: tmp[g*3+e, c] = sum_d LW[l][e][d] * h[g*3+d, c]
// One thread per (g, c): 3 reads, 9 FMA, 3 writes. Coalesced in c.
// ---------------------------------------------------------------------------
__global__ void leftmix_kernel(const float* __restrict__ h,
                               const float* __restrict__ LW,
                               float* __restrict__ tmp, int l) {
    int idx = blockIdx.x * blockDim.x + threadIdx.x;
    if (idx >= G * HC) return;
    int g = idx / HC, c = idx % HC;
    const float* base = h + (size_t)g * D * HC + c;
    float d0 = base[0], d1 = base[HC], d2 = base[2 * HC];
    const float* lw = LW + l * D * D;
    float* ob = tmp + (size_t)g * D * HC + c;
    ob[0]      = lw[0] * d0 + lw[1] * d1 + lw[2] * d2;
    ob[HC]     = lw[3] * d0 + lw[4] * d1 + lw[5] * d2;
    ob[2 * HC] = lw[6] * d0 + lw[7] * d1 + lw[8] * d2;
}

// ---------------------------------------------------------------------------
// Right channel GEMM: out[N,64] = A[N,64] @ RW[64,64]^T  (WMMA f32 16x16x4)
// ---------------------------------------------------------------------------
__global__ void right_gemm_kernel(const float* __restrict__ A,
                                  const float* __restrict__ RW,
                                  float* __restrict__ out) {
    int wid  = (int)((blockIdx.x * blockDim.x + threadIdx.x) >> 5);
    int lane = threadIdx.x & 31;
    const int mtiles = NN / 16;    // 9375
    const int ntiles = HC / 16;    // 4
    if (wid >= mtiles * ntiles) return;
    int mt = wid / ntiles, nt = wid % ntiles;
    int r0 = mt * 16, c0 = nt * 16;
    int half = lane >> 4;
    int lidx = lane & 15;

    const float* arow = A  + (size_t)(r0 + lidx) * HC;
    const float* brow = RW + (size_t)(c0 + lidx) * HC;

    v8f acc = {};
    #pragma unroll
    for (int k = 0; k < HC; k += 4) {
        v2f a = *(const v2f*)(arow + k + 2 * half);
        v2f b = *(const v2f*)(brow + k + 2 * half);
        acc = __builtin_amdgcn_wmma_f32_16x16x4_f32(
            false, a, false, b, (short)0, acc, false, false);
    }

    #pragma unroll
    for (int v = 0; v < 8; ++v) {
        int row = r0 + v + 8 * half;
        out[(size_t)row * HC + c0 + lidx] = acc[v];
    }
}

// ---------------------------------------------------------------------------
// COO SpMM: y[rows[i], :] += vals[i] * h[cols[i], :]
// One wave per nonzero: coalesced 256B row gather, 64 lane-parallel L2 atomics.
// ---------------------------------------------------------------------------
__global__ void spmm_kernel(const int* __restrict__ rows,
                            const int* __restrict__ cols,
                            const float* __restrict__ vals,
                            const float* __restrict__ h,
                            float* __restrict__ y) {
    size_t tid = (size_t)blockIdx.x * blockDim.x + threadIdx.x;
    size_t nz  = tid >> 5;
    if (nz >= (size_t)NNZ) return;
    int lane = (int)(tid & 31);
    int r = rows[nz], c = cols[nz];
    float v = vals[nz];
    const float* src = h + (size_t)c * HC;
    float* dst = y + (size_t)r * HC;
    atomicAdd(dst + lane,      v * src[lane]);
    atomicAdd(dst + lane + 32, v * src[lane + 32]);
}

// ---------------------------------------------------------------------------
// Residual/gate update: h0 = (1+tanh(eps[l,d])) * h0 - elu(y);  h = h0
// ---------------------------------------------------------------------------
__global__ void update_kernel(const float* __restrict__ y,
                              const float* __restrict__ eps,
                              float* __restrict__ h0,
                              float* __restrict__ h, int l) {
    int idx = blockIdx.x * blockDim.x + threadIdx.x;
    if (idx >= NN * HC) return;
    int node = idx / HC;
    int d = node % D;
    float gate = 1.f + tanhf(eps[l * D + d]);
    float e = eluf(y[idx]);
    float v = gate * h0[idx] - e;
    h0[idx] = v;
    h[idx]  = v;
}

// ---------------------------------------------------------------------------
// lin2: pout[G,48] = A[G,192] @ W2[40,192]^T + b2 (cols 40..47 padded w/ zeros)
// ---------------------------------------------------------------------------
__global__ void lin2_kernel(const float* __restrict__ A,
                            const float* __restrict__ W2,
                            const float* __restrict__ b2,
                            float* __restrict__ pout) {
    int wid  = (int)((blockIdx.x * blockDim.x + threadIdx.x) >> 5);
    int lane = threadIdx.x & 31;
    const int mtiles = G / 16;           // 3125
    const int ntiles = OUT_PAD / 16;     // 3
    if (wid >= mtiles * ntiles) return;
    int mt = wid / ntiles, nt = wid % ntiles;
    int r0 = mt * 16, c0 = nt * 16;
    int half = lane >> 4;
    int lidx = lane & 15;

    const float* arow = A + (size_t)(r0 + lidx) * HD;
    int wrow = c0 + lidx;
    const float* brow = W2 + (size_t)wrow * HD;
    bool bvalid = (wrow < OUT_DIM);

    v8f acc = {};
    for (int k = 0; k < HD; k += 4) {
        v2f a = *(const v2f*)(arow + k + 2 * half);
        v2f b = {0.f, 0.f};
        if (bvalid) b = *(const v2f*)(brow + k + 2 * half);
        acc = __builtin_amdgcn_wmma_f32_16x16x4_f32(
            false, a, false, b, (short)0, acc, false, false);
    }

    float bv = bvalid ? b2[wrow] : 0.f;
    #pragma unroll
    for (int v = 0; v < 8; ++v) {
        int row = r0 + v + 8 * half;
        pout[(size_t)row * OUT_PAD + c0 + lidx] = acc[v] + bv;
    }
}

// ---------------------------------------------------------------------------
// log_softmax over 40 cols; one wave per node, wave-shuffle reductions.
// ---------------------------------------------------------------------------
__global__ void logsoftmax_kernel(const float* __restrict__ pout,
                                  float* __restrict__ out) {
    int wid  = (int)((blockIdx.x * blockDim.x + threadIdx.x) >> 5);
    int lane = threadIdx.x & 31;
    if (wid >= G) return;
    const float* row = pout + (size_t)wid * OUT_PAD;
    float x1 = row[lane];                                        // lane < 40 always
    float x2 = (lane + 32 < OUT_DIM) ? row[lane + 32] : -INFINITY;
    float m = fmaxf(x1, x2);
    #pragma unroll
    for (int off = 16; off > 0; off >>= 1)
        m = fmaxf(m, __shfl_xor(m, off, 32));
    float s = __expf(x1 - m) + ((lane + 32 < OUT_DIM) ? __expf(x2 - m) : 0.f);
    #pragma unroll
    for (int off = 16; off > 0; off >>= 1)
        s += __shfl_xor(s, off, 32);
    float ls = __logf(s);
    float* o = out + (size_t)wid * OUT_DIM;
    o[lane] = x1 - m - ls;
    if (lane + 32 < OUT_DIM) o[lane + 32] = x2 - m - ls;
}

extern "C" void kernel_launch(void* const* d_in, const int* in_sizes, int n_in,
                              void* d_out, int out_size, void* d_ws, size_t ws_size,
                              hipStream_t stream) {
    const float* x        = (const float*)d_in[0];
    const int*   lap_rows = (const int*)d_in[1];
    const int*   lap_cols = (const int*)d_in[2];
    const float* lap_vals = (const float*)d_in[3];
    const float* lin1_w   = (const float*)d_in[4];
    const float* lin1_b   = (const float*)d_in[5];
    const float* left_w   = (const float*)d_in[6];
    const float* right_w  = (const float*)d_in[7];
    const float* eps      = (const float*)d_in[8];
    const float* lin2_w   = (const float*)d_in[9];
    const float* lin2_b   = (const float*)d_in[10];
    float* out = (float*)d_out;

    char* ws = (char*)d_ws;
    const size_t buf = (size_t)NN * HC * sizeof(float);   // 38.4 MB
    float* h    = (float*)(ws);
    float* h0   = (float*)(ws + buf);
    float* tmp  = (float*)(ws + 2 * buf);                 // left-mix out / SpMM acc
    float* pout = (float*)(ws + 3 * buf);                 // G*48 padded logits

    // lin1 + elu, h0 = h
    {
        int tiles  = (G / 16) * (HD / 16);                // 37500 waves
        int blocks = (tiles * 32 + 255) / 256;
        lin1_kernel<<<blocks, 256, 0, stream>>>(x, lin1_w, lin1_b, h, h0);
    }

    for (int l = 0; l < NL; ++l) {
        leftmix_kernel<<<(G * HC + 255) / 256, 256, 0, stream>>>(h, left_w, tmp, l);
        {
            int tiles  = (NN / 16) * (HC / 16);           // 37500 waves
            int blocks = (tiles * 32 + 255) / 256;
            right_gemm_kernel<<<blocks, 256, 0, stream>>>(
                tmp, right_w + (size_t)l * HC * HC, h);
        }
        hipMemsetAsync(tmp, 0, buf, stream);              // zero SpMM accumulator
        {
            long long threads = (long long)NNZ * 32;      // 1 wave / nnz
            int blocks = (int)((threads + 255) / 256);
            spmm_kernel<<<blocks, 256, 0, stream>>>(lap_rows, lap_cols, lap_vals, h, tmp);
        }
        update_kernel<<<(NN * HC + 255) / 256, 256, 0, stream>>>(tmp, eps, h0, h, l);
    }

    // lin2 (padded to 48 cols) + log_softmax
    {
        int tiles  = (G / 16) * (OUT_PAD / 16);           // 9375 waves
        int blocks = (tiles * 32 + 255) / 256;
        lin2_kernel<<<blocks, 256, 0, stream>>>(h, lin2_w, lin2_b, pout);
    }
    logsoftmax_kernel<<<(G * 32 + 255) / 256, 256, 0, stream>>>(pout, out);
}